// Net_40063454937541
// MI455X (gfx1250) — compile-verified
//
#include <hip/hip_runtime.h>
#include <math.h>

#define N_NODES 100000
#define N_EDGES 1600000
#define N_FEAT  500
#define HID     64
#define N_CLS   40

typedef float v2f __attribute__((ext_vector_type(2)));
typedef float v8f __attribute__((ext_vector_type(8)));

// ---------------------------------------------------------------------------
// v[c] = sum_j leaky_relu(ma[j]) * mb_w[j][c]   (edge MLP collapsed; bias
// cancels inside the segment softmax so it is dropped entirely)
// ---------------------------------------------------------------------------
__global__ void precompute_v(const float* __restrict__ m1a, const float* __restrict__ m1bw,
                             const float* __restrict__ m2a, const float* __restrict__ m2bw,
                             float* __restrict__ v1, float* __restrict__ v2) {
    int c = threadIdx.x;
    if (c < HID) {
        float s = 0.f;
        for (int j = 0; j < HID; ++j) {
            float g = m1a[j]; g = g > 0.f ? g : 0.2f * g;
            s += g * m1bw[j * HID + c];
        }
        v1[c] = s;
    }
    if (c < N_CLS) {
        float s = 0.f;
        for (int j = 0; j < N_CLS; ++j) {
            float g = m2a[j]; g = g > 0.f ? g : 0.2f * g;
            s += g * m2bw[j * N_CLS + c];
        }
        v2[c] = s;
    }
}

// ---------------------------------------------------------------------------
// per-source min/max of w_mul (w >= 0, IEEE bits order-preserving as uint)
// ---------------------------------------------------------------------------
__global__ void init_minmax(unsigned* __restrict__ wmaxb, unsigned* __restrict__ wminb) {
    int i = blockIdx.x * blockDim.x + threadIdx.x;
    if (i < N_NODES) { wmaxb[i] = 0u; wminb[i] = 0x7F800000u; }
}

__global__ void edge_minmax(const int* __restrict__ src, const float* __restrict__ w,
                            unsigned* __restrict__ wmaxb, unsigned* __restrict__ wminb) {
    int e = blockIdx.x * blockDim.x + threadIdx.x;
    if (e >= N_EDGES) return;
    int s = src[e];
    unsigned wb = __float_as_uint(w[e]);
    atomicMax(&wmaxb[s], wb);
    atomicMin(&wminb[s], wb);
}

// ---------------------------------------------------------------------------
// FP32 WMMA GEMM: out[N_NODES x COUT] = A[N_NODES x K] @ B[K x COUT] + bias
// One wave computes a 16-row x (NT*16)-col tile via V_WMMA_F32_16X16X4_F32.
// A frag (16x4): lanes 0-15 -> K+0,K+1 ; lanes 16-31 -> K+2,K+3 (row = lane&15)
// B frag (4x16): same K split, col = lane&15
// C/D (16x16, 8 VGPRs): vgpr j, lanes 0-15 -> row j, lanes 16-31 -> row j+8
// Padded columns (col >= COUT) use a clamped, always-valid column index so the
// loop is branch-free with EXEC all-1s; their results are never stored.
// ---------------------------------------------------------------------------
template <int K, int COUT, int NT>
__global__ __launch_bounds__(256) void gemm_wmma(const float* __restrict__ A,
                                                 const float* __restrict__ B,
                                                 const float* __restrict__ bias,
                                                 float* __restrict__ out) {
    int wid  = (blockIdx.x * blockDim.x + threadIdx.x) >> 5;
    int lane = threadIdx.x & 31;
    int row0 = wid * 16;
    if (row0 >= N_NODES) return;

    const int lo16 = lane & 15;
    const int kOff = (lane >> 4) << 1;   // 0 or 2

    v8f acc[NT];
#pragma unroll
    for (int t = 0; t < NT; ++t)
#pragma unroll
        for (int j = 0; j < 8; ++j) acc[t][j] = 0.f;

    // 32-bit element offsets (max ~50M elements) -> saddr + voffset addressing
    int aoff = (row0 + lo16) * K + kOff;
    int boff[NT];
#pragma unroll
    for (int t = 0; t < NT; ++t) {
        int n = t * 16 + lo16;
        if (n > COUT - 1) n = COUT - 1;  // clamp padded columns (branch-free)
        boff[t] = kOff * COUT + n;
    }

    for (int k0 = 0; k0 < K; k0 += 4) {
        v2f a;
        a.x = A[aoff];
        a.y = A[aoff + 1];
        v2f b[NT];
#pragma unroll
        for (int t = 0; t < NT; ++t) {
            b[t].x = B[boff[t]];
            b[t].y = B[boff[t] + COUT];
        }
#pragma unroll
        for (int t = 0; t < NT; ++t)
            acc[t] = __builtin_amdgcn_wmma_f32_16x16x4_f32(
                false, a, false, b[t], (short)0, acc[t], false, false);
        aoff += 4;
#pragma unroll
        for (int t = 0; t < NT; ++t) boff[t] += 4 * COUT;
    }

    const int rowAdd = (lane >> 4) * 8;
#pragma unroll
    for (int t = 0; t < NT; ++t) {
        int col = t * 16 + lo16;
        if (col < COUT) {
            float bb = bias[col];
#pragma unroll
            for (int j = 0; j < 8; ++j)
                out[(row0 + j + rowAdd) * COUT + col] = acc[t][j] + bb;
        }
    }
}

// ---------------------------------------------------------------------------
// segment-softmax denominator: den[s][c] += exp(w*v[c] - m[s][c])
// m[s][c] = (v[c] >= 0 ? wmax[s] : wmin[s]) * v[c]   (bias cancels)
// one thread per (edge, 4-channel group); 32-bit indexing (E*C/4 < 2^31)
// ---------------------------------------------------------------------------
template <int C>
__global__ __launch_bounds__(256) void edge_den(const int* __restrict__ src,
                                                const float* __restrict__ w,
                                                const float* __restrict__ v,
                                                const unsigned* __restrict__ wmaxb,
                                                const unsigned* __restrict__ wminb,
                                                float* __restrict__ den) {
    constexpr unsigned G = C / 4;
    unsigned idx = blockIdx.x * 256u + threadIdx.x;
    if (idx >= (unsigned)N_EDGES * G) return;
    unsigned e  = idx / G;
    unsigned cg = (idx - e * G) * 4u;
    int   s    = src[e];
    float we   = w[e];
    float wmax = __uint_as_float(wmaxb[s]);
    float wmin = __uint_as_float(wminb[s]);
    const float4 vv = *(const float4*)(v + cg);
    float vcs[4] = {vv.x, vv.y, vv.z, vv.w};
    float* dptr = den + (size_t)s * C + cg;
#pragma unroll
    for (int i = 0; i < 4; ++i) {
        float vc = vcs[i];
        float m  = (vc >= 0.f ? wmax : wmin) * vc;
        atomicAdd(dptr + i, expf(we * vc - m));
    }
}

// in-place: den -> 1/(den + 1e-16), so the message pass multiplies instead of
// dividing (6.4M rcp instead of 102M per-edge divides)
__global__ void recip_inplace(float* __restrict__ d, int n) {
    int i = blockIdx.x * blockDim.x + threadIdx.x;
    if (i < n) d[i] = 1.f / (d[i] + 1e-16f);
}

// ---------------------------------------------------------------------------
// message pass: out[d][c] += exp(w*v[c]-m) * rden[s][c] * h[s][c]
// ---------------------------------------------------------------------------
template <int C>
__global__ __launch_bounds__(256) void edge_msg(const int* __restrict__ src,
                                                const int* __restrict__ dst,
                                                const float* __restrict__ w,
                                                const float* __restrict__ v,
                                                const unsigned* __restrict__ wmaxb,
                                                const unsigned* __restrict__ wminb,
                                                const float* __restrict__ rden,
                                                const float* __restrict__ h,
                                                float* __restrict__ out) {
    constexpr unsigned G = C / 4;
    unsigned idx = blockIdx.x * 256u + threadIdx.x;
    if (idx >= (unsigned)N_EDGES * G) return;
    unsigned e  = idx / G;
    unsigned cg = (idx - e * G) * 4u;
    int   s    = src[e];
    int   d    = dst[e];
    float we   = w[e];
    float wmax = __uint_as_float(wmaxb[s]);
    float wmin = __uint_as_float(wminb[s]);
    const float4 vv = *(const float4*)(v + cg);
    const float4 rd = *(const float4*)(rden + (size_t)s * C + cg);
    const float4 hh = *(const float4*)(h + (size_t)s * C + cg);
    float vcs[4] = {vv.x, vv.y, vv.z, vv.w};
    float rds[4] = {rd.x, rd.y, rd.z, rd.w};
    float hhs[4] = {hh.x, hh.y, hh.z, hh.w};
    float* optr = out + (size_t)d * C + cg;
#pragma unroll
    for (int i = 0; i < 4; ++i) {
        float vc    = vcs[i];
        float m     = (vc >= 0.f ? wmax : wmin) * vc;
        float alpha = expf(we * vc - m) * rds[i];
        atomicAdd(optr + i, alpha * hhs[i]);
    }
}

__global__ void elu_inplace(float* __restrict__ a, int n) {
    int i = blockIdx.x * blockDim.x + threadIdx.x;
    if (i < n) { float x = a[i]; a[i] = x > 0.f ? x : expm1f(x); }
}

__global__ void logsoftmax40(float* __restrict__ out) {
    int n = blockIdx.x * blockDim.x + threadIdx.x;
    if (n >= N_NODES) return;
    float* p = out + (size_t)n * N_CLS;
    float m = -INFINITY;
#pragma unroll
    for (int c = 0; c < N_CLS; ++c) m = fmaxf(m, p[c]);
    float s = 0.f;
#pragma unroll
    for (int c = 0; c < N_CLS; ++c) s += expf(p[c] - m);
    float l = m + logf(s);
#pragma unroll
    for (int c = 0; c < N_CLS; ++c) p[c] = p[c] - l;
}

// ---------------------------------------------------------------------------
extern "C" void kernel_launch(void* const* d_in, const int* in_sizes, int n_in,
                              void* d_out, int out_size, void* d_ws, size_t ws_size,
                              hipStream_t stream) {
    const float* x    = (const float*)d_in[0];
    const int*   ei   = (const int*)  d_in[1];
    const float* wmul = (const float*)d_in[2];
    const float* w1   = (const float*)d_in[3];
    const float* b1   = (const float*)d_in[4];
    const float* m1a  = (const float*)d_in[5];
    const float* m1bw = (const float*)d_in[6];
    // d_in[7] = m1b_b : cancels in segment softmax
    const float* w2   = (const float*)d_in[8];
    const float* b2   = (const float*)d_in[9];
    const float* m2a  = (const float*)d_in[10];
    const float* m2bw = (const float*)d_in[11];
    // d_in[12] = m2b_b : cancels

    const int* src = ei;
    const int* dst = ei + N_EDGES;

    float*    ws    = (float*)d_ws;
    float*    v1    = ws;                                   // 64
    float*    v2    = ws + 64;                              // 64
    unsigned* wmaxb = (unsigned*)(ws + 128);                // N
    unsigned* wminb = wmaxb + N_NODES;                      // N
    float*    h1    = (float*)(wminb + N_NODES);            // N*64 (reused as h2)
    float*    den1  = h1   + (size_t)N_NODES * HID;         // N*64 (reused as den2)
    float*    agg1  = den1 + (size_t)N_NODES * HID;         // N*64 -> elu -> a

    hipMemsetAsync(den1, 0, (size_t)N_NODES * HID * sizeof(float), stream);
    hipMemsetAsync(agg1, 0, (size_t)N_NODES * HID * sizeof(float), stream);
    hipMemsetAsync(d_out, 0, (size_t)N_NODES * N_CLS * sizeof(float), stream);

    precompute_v<<<1, 64, 0, stream>>>(m1a, m1bw, m2a, m2bw, v1, v2);
    init_minmax<<<(N_NODES + 255) / 256, 256, 0, stream>>>(wmaxb, wminb);
    edge_minmax<<<(N_EDGES + 255) / 256, 256, 0, stream>>>(src, wmul, wmaxb, wminb);

    const int waves   = N_NODES / 16;                 // 6250 (exact)
    const int gemmBlk = (waves * 32 + 255) / 256;

    // ---- layer 1 ----
    gemm_wmma<N_FEAT, HID, 4><<<gemmBlk, 256, 0, stream>>>(x, w1, b1, h1);

    unsigned g1 = ((unsigned)N_EDGES * (HID / 4) + 255u) / 256u;
    edge_den<HID><<<g1, 256, 0, stream>>>(src, wmul, v1, wmaxb, wminb, den1);
    recip_inplace<<<(N_NODES * HID + 255) / 256, 256, 0, stream>>>(den1, N_NODES * HID);
    edge_msg<HID><<<g1, 256, 0, stream>>>(src, dst, wmul, v1, wmaxb, wminb, den1, h1, agg1);
    elu_inplace<<<(N_NODES * HID + 255) / 256, 256, 0, stream>>>(agg1, N_NODES * HID);

    // ---- layer 2 (reuse h1/den1 regions) ----
    float* h2   = h1;
    float* den2 = den1;
    gemm_wmma<HID, N_CLS, 3><<<gemmBlk, 256, 0, stream>>>(agg1, w2, b2, h2);
    hipMemsetAsync(den2, 0, (size_t)N_NODES * N_CLS * sizeof(float), stream);

    unsigned g2 = ((unsigned)N_EDGES * (N_CLS / 4) + 255u) / 256u;
    edge_den<N_CLS><<<g2, 256, 0, stream>>>(src, wmul, v2, wmaxb, wminb, den2);
    recip_inplace<<<(N_NODES * N_CLS + 255) / 256, 256, 0, stream>>>(den2, N_NODES * N_CLS);
    edge_msg<N_CLS><<<g2, 256, 0, stream>>>(src, dst, wmul, v2, wmaxb, wminb, den2, h2,
                                            (float*)d_out);
    logsoftmax40<<<(N_NODES + 255) / 256, 256, 0, stream>>>((float*)d_out);
}